// EdgeFeatureEncoding_21492016349937
// MI455X (gfx1250) — compile-verified
//
#include <hip/hip_runtime.h>

typedef __attribute__((ext_vector_type(2))) float v2f;
typedef __attribute__((ext_vector_type(4))) float v4f;
typedef __attribute__((ext_vector_type(8))) float v8f;

#define EDGE_FEAT 128
#define N_HEADS 8
#define NPAD 16   // heads padded to 16 (WMMA N dim)

// -------- Kernel 1: stream-zero the 512MB output with NT b128 stores --------
__global__ void __launch_bounds__(256)
zero_out_kernel(v4f* __restrict__ out, long long n4) {
    long long idx    = (long long)blockIdx.x * blockDim.x + threadIdx.x;
    long long stride = (long long)gridDim.x * blockDim.x;
    v4f z = {0.f, 0.f, 0.f, 0.f};
    for (long long i = idx; i < n4; i += stride) {
        __builtin_nontemporal_store(z, &out[i]);
    }
}

// -------- Kernel 2: WMMA f32 16x16x4 projection + atomic scatter-add --------
// One wave handles 16 edges x 16 heads (heads 8..15 are zero padding).
// A (16x4 f32): lanes 0-15 hold K=k0,k0+1 ; lanes 16-31 hold K=k0+2,k0+3 (M = lane&15)
// B (4x16 f32): mirrored with N = lane&15
// C/D (16x16 f32): VGPR v -> M = v + 8*(lane>=16), N = lane&15
//
// LDS W layout: sW2[k/2][n] = (W[k][n], W[k+1][n]), n padded to 16 with zeros,
// so each lane fetches its two B values with ONE unconditional ds_load_b64.
__global__ void __launch_bounds__(256)
edge_encode_kernel(const int* __restrict__ edge_index,
                   const float* __restrict__ edge_attr,
                   const int* __restrict__ num_nodes_p,
                   const float* __restrict__ Wg,
                   const float* __restrict__ bg,
                   float* __restrict__ out,
                   int E) {
    __shared__ v2f   sW2[(EDGE_FEAT / 2) * NPAD];   // 64 K-pairs x 16 heads = 8KB
    __shared__ float sB[N_HEADS];

    const int tid = threadIdx.x;
    // Stage W as zero-padded K-pairs: 1024 entries, 4 per thread
    #pragma unroll
    for (int t = tid; t < (EDGE_FEAT / 2) * NPAD; t += 256) {
        const int k2 = t >> 4;        // K pair index (k = 2*k2, 2*k2+1)
        const int n  = t & 15;
        v2f val = {0.f, 0.f};
        if (n < N_HEADS) {
            val.x = Wg[(2 * k2)     * N_HEADS + n];
            val.y = Wg[(2 * k2 + 1) * N_HEADS + n];
        }
        sW2[t] = val;
    }
    if (tid < N_HEADS) sB[tid] = bg[tid];
    __syncthreads();

    const int N    = num_nodes_p[0];
    const int lane = tid & 31;
    const int wave = tid >> 5;
    const int n    = lane & 15;              // head column (0..15, >=8 is pad)
    const int hi   = lane >> 4;              // 0 for lanes 0-15, 1 for 16-31
    const int kSel = hi << 1;                // K sub-offset: 0 or 2

    const int edgeBase = blockIdx.x * 128 + wave * 16;

    // A row pointer for this lane (clamped for tail safety; tail lanes are
    // excluded from the scatter below, so garbage accumulation is harmless).
    int eA   = edgeBase + (lane & 15);
    int rowA = (eA < E) ? eA : 0;
    const float* arow = edge_attr + (long long)rowA * EDGE_FEAT + kSel;

    // B pointer: pair index p = (k0 + kSel)/2, entry = p*16 + n
    const v2f* brow = sW2 + (kSel >> 1) * NPAD + n;

    v8f c = {};
    #pragma unroll 8
    for (int k0 = 0; k0 < EDGE_FEAT; k0 += 4) {
        v2f a  = *(const v2f*)(arow + k0);       // 8B-aligned (k0+kSel even)
        v2f bb = brow[(k0 >> 1) * NPAD];         // one ds_load_b64, no branches
        // 8 args: (neg_a, A, neg_b, B, c_mod, C, reuse_a, reuse_b)
        c = __builtin_amdgcn_wmma_f32_16x16x4_f32(
                false, a, false, bb, (short)0, c, false, false);
    }

    // Scatter-add: each (m, n<8) pair lives in exactly one lane/VGPR slot.
    if (n < N_HEADS) {
        const float biasn = sB[n];
        const int mAdd = hi << 3;                // rows 0-7 or 8-15
        #pragma unroll
        for (int v = 0; v < 8; ++v) {
            const int e = edgeBase + v + mAdd;
            if (e < E) {
                const int i = edge_index[e];
                const int j = edge_index[E + e];
                if (i >= 0 && i < N && j >= 0 && j < N) {
                    const size_t off = (((size_t)i * (size_t)N) + (size_t)j)
                                       * (size_t)N_HEADS + (size_t)n;
                    unsafeAtomicAdd(out + off, c[v] + biasn);
                }
            }
        }
    }
}

extern "C" void kernel_launch(void* const* d_in, const int* in_sizes, int n_in,
                              void* d_out, int out_size, void* d_ws, size_t ws_size,
                              hipStream_t stream) {
    const int*   edge_index = (const int*)d_in[0];
    const float* edge_attr  = (const float*)d_in[1];
    const int*   num_nodes  = (const int*)d_in[2];
    const float* W          = (const float*)d_in[3];
    const float* b          = (const float*)d_in[4];
    float*       out        = (float*)d_out;

    const int E = in_sizes[0] / 2;

    // Zero the dense (N, N, 8) output: out_size/4 float4 stores
    const long long n4 = (long long)out_size / 4;
    zero_out_kernel<<<4096, 256, 0, stream>>>((v4f*)out, n4);

    // 8 waves/block, 16 edges/wave -> 128 edges per block
    const int blocks = (E + 127) / 128;
    edge_encode_kernel<<<blocks, 256, 0, stream>>>(
        edge_index, edge_attr, num_nodes, W, b, out, E);
}